// anim_conv_5025111736623
// MI455X (gfx1250) — compile-verified
//
#include <hip/hip_runtime.h>
#include <hip/hip_bf16.h>
#include <math.h>

// ---------------- problem constants (match reference) ----------------
#define BGRAPHS 64
#define VNODES  256
#define NNODES  (BGRAPHS * VNODES)     // 16384
#define DEGAVG  32
#define NEDGES  (NNODES * DEGAVG)      // 524288
#define CH      24
#define EOUT    6
#define BN_EPS  1e-5f

typedef float v2f __attribute__((ext_vector_type(2)));
typedef float v8f __attribute__((ext_vector_type(8)));

struct Layer {
  const float* W;     // [in_dim, out_dim] row-major
  const float* b;
  const float* beta;
  const float* gamma;
  const float* mean;
  const float* var;
};
struct Mlp3 { Layer L[3]; };

// ---- stage one layer's weights (zero-padded KPADxNPAD) + fused BN coeffs ----
template <int IN, int OUT>
__device__ __forceinline__ void stage_layer(const Layer& L, float* __restrict__ wlds,
                                            float* __restrict__ sclds,
                                            float* __restrict__ shlds, int tid) {
  constexpr int KPAD = ((IN + 3) / 4) * 4;
  constexpr int NPAD = ((OUT + 15) / 16) * 16;
  for (int idx = tid; idx < KPAD * NPAD; idx += 256) {
    const int k = idx / NPAD, n = idx % NPAD;
    wlds[idx] = (k < IN && n < OUT) ? L.W[k * OUT + n] : 0.f;
  }
  for (int n = tid; n < NPAD; n += 256) {
    float sc = 0.f, sh = 0.f;
    if (n < OUT) {
      sc = L.gamma[n] * rsqrtf(L.var[n] + BN_EPS);
      sh = sc * (L.b[n] - L.mean[n]) + L.beta[n];
    }
    sclds[n] = sc;
    shlds[n] = sh;
  }
}

// ---------------------------------------------------------------------
// One MLP layer on a wave's 16-row tile, LDS -> LDS, fp32 WMMA.
// A frag: lane = M%16, half selects K{0,1}/{2,3} across the 2 VGPRs.
// B frag: lane = N%16, half selects K block (mirror of A).
// C/D  : VGPR r, lane L -> (M = r + 8*(L>=16), N = L%16).
// Fully unrolled; all operand loads are unconditional LDS loads.
// ---------------------------------------------------------------------
template <int LDSW, int IN, int OUT>
__device__ __forceinline__ void mlp_layer_tile(const float (*src)[LDSW],
                                               float (*dst)[LDSW],
                                               const float* __restrict__ wlds,
                                               const float* __restrict__ sclds,
                                               const float* __restrict__ shlds,
                                               int r0, int lane) {
  constexpr int KSTEPS = (IN + 3) / 4;
  constexpr int NTILES = (OUT + 15) / 16;
  constexpr int NPAD   = NTILES * 16;
  const int m     = lane & 15;
  const int n     = lane & 15;
  const int kb    = (lane >> 4) << 1;   // 0 or 2
  const int mhalf = (lane >> 4) << 3;   // 0 or 8

#pragma unroll
  for (int nt = 0; nt < NTILES; ++nt) {
    const int nn = (nt << 4) + n;
    v8f acc = {0.f, 0.f, 0.f, 0.f, 0.f, 0.f, 0.f, 0.f};
#pragma unroll
    for (int s = 0; s < KSTEPS; ++s) {
      const int k0 = (s << 2) + kb;
      v2f a, bfrag;
      a.x     = src[r0 + m][k0];
      a.y     = src[r0 + m][k0 + 1];
      bfrag.x = wlds[k0 * NPAD + nn];
      bfrag.y = wlds[(k0 + 1) * NPAD + nn];
      acc = __builtin_amdgcn_wmma_f32_16x16x4_f32(false, a, false, bfrag,
                                                  (short)0, acc, false, false);
    }
    const float sc = sclds[nn];
    const float sh = shlds[nn];
    if (nn < LDSW) {   // only last tile can exceed; padded cols store 0 (sc=sh=0)
#pragma unroll
      for (int r = 0; r < 8; ++r) {
        dst[r0 + mhalf + r][nn] = fmaxf(sc * acc[r] + sh, 0.f);
      }
    }
  }
}

// ---------------- edge MLP: [E,6] -> [E,6] + mean -> ew[E] ----------------
#define ELDSW 17
__global__ __launch_bounds__(256) void edge_mlp_kernel(
    const float* __restrict__ attr, Mlp3 p,
    float* __restrict__ ef, float* __restrict__ ew) {
  __shared__ float bufA[128][ELDSW];
  __shared__ float bufB[128][ELDSW];
  __shared__ float w0[8 * 16], w1[16 * 16], w2[16 * 16];
  __shared__ float sc0[16], sh0[16], sc1[16], sh1[16], sc2[16], sh2[16];

  const int tid  = threadIdx.x;
  const int lane = tid & 31;
  const int wave = tid >> 5;
  const int r0   = wave << 4;
  const int eBase = blockIdx.x * 128 + r0;

  stage_layer<6, 14>(p.L[0], w0, sc0, sh0, tid);
  stage_layer<14, 14>(p.L[1], w1, sc1, sh1, tid);
  stage_layer<14, 6>(p.L[2], w2, sc2, sh2, tid);

  { // stage 16 edges x 16 cols (zero-padded past col 5)
    const int row = lane & 15;
    if (lane < 16) {
      const int e = eBase + row;
#pragma unroll
      for (int c = 0; c < 6; ++c) bufA[r0 + row][c] = attr[e * 6 + c];
    } else {
#pragma unroll
      for (int c = 6; c < 16; ++c) bufA[r0 + row][c] = 0.f;
    }
  }
  __syncthreads();
  mlp_layer_tile<ELDSW, 6, 14>(bufA, bufB, w0, sc0, sh0, r0, lane);
  __syncthreads();
  mlp_layer_tile<ELDSW, 14, 14>(bufB, bufA, w1, sc1, sh1, r0, lane);
  __syncthreads();
  mlp_layer_tile<ELDSW, 14, 6>(bufA, bufB, w2, sc2, sh2, r0, lane);
  __syncthreads();
  if (lane < 16) {
    const int e = eBase + lane;
    float s = 0.f;
#pragma unroll
    for (int c = 0; c < 6; ++c) {
      const float v = bufB[r0 + lane][c];
      ef[e * 6 + c] = v;
      s += v;
    }
    ew[e] = s * (1.f / 6.f);
  }
}

// ---------------- node MLP: [N,24] -> [N,24] ----------------
#define NLDSW 37
__global__ __launch_bounds__(256) void node_mlp_kernel(
    const float* __restrict__ x, Mlp3 p, float* __restrict__ h) {
  __shared__ float bufA[128][NLDSW];
  __shared__ float bufB[128][NLDSW];
  __shared__ float w0[24 * 48], w1[36 * 48], w2[36 * 32];
  __shared__ float sc0[48], sh0[48], sc1[48], sh1[48], sc2[32], sh2[32];

  const int tid  = threadIdx.x;
  const int lane = tid & 31;
  const int wave = tid >> 5;
  const int r0   = wave << 4;
  const int nBase = blockIdx.x * 128 + r0;

  stage_layer<24, 36>(p.L[0], w0, sc0, sh0, tid);
  stage_layer<36, 36>(p.L[1], w1, sc1, sh1, tid);
  stage_layer<36, 24>(p.L[2], w2, sc2, sh2, tid);

  { // stage 16 nodes x 24 cols
    const int row = lane & 15;
    const int nd = nBase + row;
    const int cbase = (lane >> 4) * 12;
#pragma unroll
    for (int j = 0; j < 12; ++j) {
      const int c = cbase + j;
      bufA[r0 + row][c] = x[nd * CH + c];
    }
  }
  __syncthreads();
  mlp_layer_tile<NLDSW, 24, 36>(bufA, bufB, w0, sc0, sh0, r0, lane);
  __syncthreads();
  mlp_layer_tile<NLDSW, 36, 36>(bufB, bufA, w1, sc1, sh1, r0, lane);
  __syncthreads();
  mlp_layer_tile<NLDSW, 36, 24>(bufA, bufB, w2, sc2, sh2, r0, lane);
  __syncthreads();
  {
    const int row = lane & 15;
    const int nd = nBase + row;
    const int cbase = (lane >> 4) * 12;
#pragma unroll
    for (int j = 0; j < 12; ++j) {
      const int c = cbase + j;
      h[nd * CH + c] = bufB[r0 + row][c];
    }
  }
}

// ---------------- GGC per-layer m = h @ W[i]  (24x24, fp32 WMMA) ----------------
__global__ __launch_bounds__(256) void ggc_matmul_kernel(
    const float* __restrict__ h, const float* __restrict__ W,
    float* __restrict__ m_out) {
  __shared__ float buf[128][25];
  __shared__ float wl[24 * 32];
  const int tid  = threadIdx.x;
  const int lane = tid & 31;
  const int wave = tid >> 5;
  const int r0   = wave << 4;
  const int nBase = blockIdx.x * 128 + r0;

  for (int idx = tid; idx < 24 * 32; idx += 256) {
    const int k = idx >> 5, n = idx & 31;
    wl[idx] = (n < CH) ? W[k * CH + n] : 0.f;
  }
  {
    const int row = lane & 15;
    const int nd = nBase + row;
    const int cbase = (lane >> 4) * 12;
#pragma unroll
    for (int j = 0; j < 12; ++j) {
      const int c = cbase + j;
      buf[r0 + row][c] = h[nd * CH + c];
    }
  }
  __syncthreads();

  const int mm    = lane & 15;
  const int n     = lane & 15;
  const int kb    = (lane >> 4) << 1;
  const int mhalf = (lane >> 4) << 3;
#pragma unroll
  for (int nt = 0; nt < 2; ++nt) {
    const int nn = (nt << 4) + n;
    v8f acc = {0.f, 0.f, 0.f, 0.f, 0.f, 0.f, 0.f, 0.f};
#pragma unroll
    for (int s = 0; s < 6; ++s) {   // K = 24 -> 6 steps of 4
      const int k0 = (s << 2) + kb;
      v2f a, bfrag;
      a.x     = buf[r0 + mm][k0];
      a.y     = buf[r0 + mm][k0 + 1];
      bfrag.x = wl[(k0 << 5) + nn];
      bfrag.y = wl[((k0 + 1) << 5) + nn];
      acc = __builtin_amdgcn_wmma_f32_16x16x4_f32(false, a, false, bfrag,
                                                  (short)0, acc, false, false);
    }
    if (nn < CH) {
#pragma unroll
      for (int r = 0; r < 8; ++r)
        m_out[(nBase + mhalf + r) * CH + nn] = acc[r];
    }
  }
}

// ---------------- misc kernels ----------------
__global__ void zero_kernel(float* __restrict__ p, int n) {
  const int i = blockIdx.x * blockDim.x + threadIdx.x;
  if (i < n) p[i] = 0.f;
}

// msg scatter: agg[dst] += ew[e] * m[src]   (m rows are 96B = 16B aligned)
__global__ void scatter_msg_kernel(const int* __restrict__ src,
                                   const int* __restrict__ dst,
                                   const float* __restrict__ ew,
                                   const float* __restrict__ m,
                                   float* __restrict__ agg, int Etot) {
  const int e = blockIdx.x * blockDim.x + threadIdx.x;
  if (e >= Etot) return;
  const int s = src[e], d = dst[e];
  const float w = ew[e];
  const float4* mr4 = (const float4*)(m + s * CH);
  float* ag = agg + d * CH;
#pragma unroll
  for (int q = 0; q < 6; ++q) {
    const float4 v = mr4[q];
    atomicAdd(&ag[q * 4 + 0], w * v.x);
    atomicAdd(&ag[q * 4 + 1], w * v.y);
    atomicAdd(&ag[q * 4 + 2], w * v.z);
    atomicAdd(&ag[q * 4 + 3], w * v.w);
  }
}

// GRU update per node (weights are wave-uniform -> scalar-cached)
__global__ __launch_bounds__(256) void gru_kernel(
    const float* __restrict__ agg, float* __restrict__ h,
    const float* __restrict__ w_ih, const float* __restrict__ w_hh,
    const float* __restrict__ b_ih, const float* __restrict__ b_hh, int Ntot) {
  const int i = blockIdx.x * blockDim.x + threadIdx.x;
  if (i >= Ntot) return;
  float hr[CH], ar[CH], hn[CH];
  for (int k = 0; k < CH; ++k) { hr[k] = h[i * CH + k]; ar[k] = agg[i * CH + k]; }
  for (int j = 0; j < CH; ++j) {
    float gir = b_ih[j],          ghr = b_hh[j];
    float giz = b_ih[CH + j],     ghz = b_hh[CH + j];
    float gin = b_ih[2 * CH + j], ghn = b_hh[2 * CH + j];
    for (int k = 0; k < CH; ++k) {
      gir += w_ih[ j           * CH + k] * ar[k];
      giz += w_ih[(CH + j)     * CH + k] * ar[k];
      gin += w_ih[(2 * CH + j) * CH + k] * ar[k];
      ghr += w_hh[ j           * CH + k] * hr[k];
      ghz += w_hh[(CH + j)     * CH + k] * hr[k];
      ghn += w_hh[(2 * CH + j) * CH + k] * hr[k];
    }
    const float r = 1.f / (1.f + expf(-(gir + ghr)));
    const float z = 1.f / (1.f + expf(-(giz + ghz)));
    const float nv = tanhf(gin + r * ghn);
    hn[j] = (1.f - z) * nv + z * hr[j];
  }
  for (int j = 0; j < CH; ++j) h[i * CH + j] = hn[j];
}

// final einsum as per-edge scatter (A never materialized):
// out[b,c,t,dst%V] += ef[e,c] * relu(h[src, c*4+t]),  b = src/V
__global__ void out_scatter_kernel(const int* __restrict__ src,
                                   const int* __restrict__ dst,
                                   const float* __restrict__ ef,
                                   const float* __restrict__ h,
                                   float* __restrict__ out, int Etot) {
  const int e = blockIdx.x * blockDim.x + threadIdx.x;
  if (e >= Etot) return;
  const int s = src[e], d = dst[e];
  const int b = s >> 8;            // /V (V==256)
  const int w = d & 255;           // %V
  const float* efr = ef + e * EOUT;
  const float* hrow = h + s * CH;
  float* ob = out + b * (EOUT * 4 * VNODES);
#pragma unroll
  for (int c = 0; c < EOUT; ++c) {
    const float fc = efr[c];
#pragma unroll
    for (int t = 0; t < 4; ++t) {
      const float hv = fmaxf(hrow[(c << 2) + t], 0.f);
      atomicAdd(&ob[((c << 2) + t) * VNODES + w], fc * hv);
    }
  }
}

// ---------------------------------------------------------------------
extern "C" void kernel_launch(void* const* d_in, const int* in_sizes, int n_in,
                              void* d_out, int out_size, void* d_ws, size_t ws_size,
                              hipStream_t stream) {
  // Input flattening: top-level dict insertion order, params as JAX pytree
  // leaves (dict keys sorted alphabetically, lists in order):
  //  0 x[N,24]  1 edge_attr[E,6]  2 edge_index[2,E]  3 batch[N]
  //  params.edge_mlp layers 0..2: (W,b,beta,gamma,mean,var) -> 4..21
  //  params.ggc: W(22) b_hh(23) b_ih(24) w_hh(25) w_ih(26)
  //  params.node_mlp layers 0..2: -> 27..44
  const float* x         = (const float*)d_in[0];
  const float* edge_attr = (const float*)d_in[1];
  const int*   eidx      = (const int*)d_in[2];
  const int*   esrc      = eidx;
  const int*   edst      = eidx + NEDGES;

  auto mkLayer = [&](int base) {
    Layer L;
    L.W     = (const float*)d_in[base + 0];
    L.b     = (const float*)d_in[base + 1];
    L.beta  = (const float*)d_in[base + 2];
    L.gamma = (const float*)d_in[base + 3];
    L.mean  = (const float*)d_in[base + 4];
    L.var   = (const float*)d_in[base + 5];
    return L;
  };
  Mlp3 edgeP, nodeP;
  edgeP.L[0] = mkLayer(4);
  edgeP.L[1] = mkLayer(10);
  edgeP.L[2] = mkLayer(16);
  const float* ggc_W    = (const float*)d_in[22];  // [3,24,24]
  const float* ggc_b_hh = (const float*)d_in[23];
  const float* ggc_b_ih = (const float*)d_in[24];
  const float* ggc_w_hh = (const float*)d_in[25];
  const float* ggc_w_ih = (const float*)d_in[26];
  nodeP.L[0] = mkLayer(27);
  nodeP.L[1] = mkLayer(33);
  nodeP.L[2] = mkLayer(39);

  // workspace carve-up (all sizes 256B-aligned); total ~18.5 MB
  char* ws = (char*)d_ws;
  float* ef  = (float*)(ws);                                   // E*6
  float* ew  = (float*)(ws + (size_t)NEDGES * 6 * 4);          // E
  float* h   = (float*)(ws + (size_t)NEDGES * 7 * 4);          // N*24
  float* m   = (float*)(ws + (size_t)NEDGES * 7 * 4 + (size_t)NNODES * CH * 4);
  float* agg = (float*)(ws + (size_t)NEDGES * 7 * 4 + (size_t)NNODES * CH * 8);
  float* out = (float*)d_out;

  // 1) edge MLP -> ef, ew        (E/128 = 4096 blocks, 8 waves each)
  edge_mlp_kernel<<<NEDGES / 128, 256, 0, stream>>>(edge_attr, edgeP, ef, ew);
  // 2) node MLP -> h
  node_mlp_kernel<<<NNODES / 128, 256, 0, stream>>>(x, nodeP, h);
  // 3) three GatedGraphConv layers
  for (int i = 0; i < 3; ++i) {
    ggc_matmul_kernel<<<NNODES / 128, 256, 0, stream>>>(h, ggc_W + i * CH * CH, m);
    zero_kernel<<<(NNODES * CH + 255) / 256, 256, 0, stream>>>(agg, NNODES * CH);
    scatter_msg_kernel<<<NEDGES / 256, 256, 0, stream>>>(esrc, edst, ew, m, agg, NEDGES);
    gru_kernel<<<NNODES / 256, 256, 0, stream>>>(agg, h, ggc_w_ih, ggc_w_hh,
                                                 ggc_b_ih, ggc_b_hh, NNODES);
  }
  // 4) out = einsum(hf, Ap) via per-edge scatter; zero output first
  zero_kernel<<<(out_size + 255) / 256, 256, 0, stream>>>(out, out_size);
  out_scatter_kernel<<<NEDGES / 256, 256, 0, stream>>>(esrc, edst, ef, h, out, NEDGES);
}